// NT_Xent_15350213116619
// MI455X (gfx1250) — compile-verified
//
#include <hip/hip_runtime.h>
#include <math.h>

// CDNA5 / gfx1250: wave32, WMMA (not MFMA).
typedef float v2f __attribute__((ext_vector_type(2)));
typedef float v8f __attribute__((ext_vector_type(8)));

#define B_ROWS 8192
#define NFEAT  128
#define INV_TAU 2.0f             // 1/tau, tau = 0.5
#define LOG2E_X2 2.885390081777927f  // 2*log2(e): exp(x/tau) = exp2(x * this)
#define APITCH 132               // LDS row pitch: 132 % 64 = 4 -> bank-conflict-free, 8B aligned
#define MROWS  32                // rows per workgroup (two 16-row WMMA tiles share each B panel)

// ---------------------------------------------------------------------------
// Kernel 1: row L2-normalization. One wave per row, float4 per lane.
// ---------------------------------------------------------------------------
__global__ void __launch_bounds__(256) nrm_kernel(const float* __restrict__ z,
                                                  float* __restrict__ zn) {
  const int w    = threadIdx.x >> 5;
  const int lane = threadIdx.x & 31;
  const int row  = blockIdx.x * 8 + w;
  const float4* src = (const float4*)(z + (size_t)row * NFEAT);
  float4 v = src[lane];
  float ss = v.x * v.x + v.y * v.y + v.z * v.z + v.w * v.w;
#pragma unroll
  for (int m = 16; m >= 1; m >>= 1) ss += __shfl_xor(ss, m, 32);
  const float inv = 1.0f / sqrtf(fmaxf(ss, 1e-12f));
  float4 o;
  o.x = v.x * inv; o.y = v.y * inv; o.z = v.z * inv; o.w = v.w * inv;
  ((float4*)(zn + (size_t)row * NFEAT))[lane] = o;
}

// ---------------------------------------------------------------------------
// Kernel 2: fused Gram-row-block + sum-exp + diagonal/pos capture.
// 32 rows x 8192 cols per workgroup; two 16-row A tiles (LDS) share every
// B panel (registers). B is prefetched as a 32 x v2f batch so the loads issue
// as one clause; A LDS indices are laundered through asm() per tile to stop
// LICM from hoisting 128 A values into registers (which starved B of regs).
// Hot epilogue: unconditional s += exp2(c * 2log2e); the self term is
// subtracted at the end using sDiag captured in the (rare, wave-uniform)
// diagonal tiles. Logits bounded by |cos|/tau <= 2, so no running max needed.
//
// f32 WMMA per-lane layouts (ISA 7.12.2):
//   A 16x4 : lanes 0-15 -> M=lane,  K = 4kt+{0,1}; lanes 16-31 -> K = 4kt+{2,3}
//   B 4x16 : lanes 0-15 -> N=lane,  K = 4kt+{0,1}; lanes 16-31 -> K = 4kt+{2,3}
//   C 16x16: VGPR r -> M = r + 8*(lane>=16), N = lane&15
// ---------------------------------------------------------------------------
__global__ void __launch_bounds__(256) ntx_kernel(const float* __restrict__ zn,
                                                  float* __restrict__ partial) {
  __shared__ float sA[MROWS * APITCH];
  __shared__ float sS[8][MROWS];
  __shared__ float sPos[MROWS];
  __shared__ float sDiag[MROWS];

  const int tid  = threadIdx.x;
  const int w    = tid >> 5;
  const int lane = tid & 31;
  const int l16  = lane & 15;
  const int hi   = lane >> 4;
  const int row0 = blockIdx.x * MROWS;
  const int diag0 = blockIdx.x * 2;     // first of the two diagonal column tiles

  // Stage the 32 A-rows (32x128 f32 = 16 KB) into LDS, coalesced.
#pragma unroll
  for (int i = 0; i < 16; ++i) {
    const int idx = tid + 256 * i;
    const int r = idx >> 7, k = idx & 127;
    sA[r * APITCH + k] = zn[(size_t)(row0 + r) * NFEAT + k];
  }
  __syncthreads();

  float s0[8], s1[8];
#pragma unroll
  for (int r = 0; r < 8; ++r) { s0[r] = 0.0f; s1[r] = 0.0f; }

  const unsigned io0base = (unsigned)(l16 * APITCH + 2 * hi);          // rows 0..15
  const unsigned io1base = io0base + 16u * APITCH;                     // rows 16..31

  for (int t = 0; t < 64; ++t) {
    const int ct  = w + 8 * t;          // column tile
    const int col = ct * 16 + l16;      // this lane's B column (global row of zn)
    const float* bptr = zn + (size_t)col * NFEAT + 2 * hi;

    // Launder A indices: keeps the ds_loads inside this iteration (bounded
    // register lifetime) instead of 128 hoisted live-forever A registers.
    unsigned io0 = io0base, io1 = io1base;
    asm volatile("" : "+v"(io0), "+v"(io1));

    // Batch-issue the 32 B loads (one clause; latency amortized).
    v2f breg[32];
#pragma unroll
    for (int kt = 0; kt < 32; ++kt)
      breg[kt] = *(const v2f*)(bptr + 4 * kt);

    v8f c0 = {0.f, 0.f, 0.f, 0.f, 0.f, 0.f, 0.f, 0.f};
    v8f c1 = {0.f, 0.f, 0.f, 0.f, 0.f, 0.f, 0.f, 0.f};
#pragma unroll
    for (int kt = 0; kt < 32; ++kt) {
      v2f a0 = *(const v2f*)(&sA[io0 + 4 * kt]);
      v2f a1 = *(const v2f*)(&sA[io1 + 4 * kt]);
      c0 = __builtin_amdgcn_wmma_f32_16x16x4_f32(false, a0, false, breg[kt],
                                                 (short)0, c0, false, false);
      c1 = __builtin_amdgcn_wmma_f32_16x16x4_f32(false, a1, false, breg[kt],
                                                 (short)0, c1, false, false);
    }

    // Hot epilogue: 1 v_exp + 1 add per element, no masking.
#pragma unroll
    for (int r = 0; r < 8; ++r) {
      s0[r] += exp2f(c0[r] * LOG2E_X2);
      s1[r] += exp2f(c1[r] * LOG2E_X2);
    }

    // Rare (wave-uniform) path: this tile holds the diagonal 32x16 strip.
    if ((unsigned)(ct - diag0) < 2u) {
#pragma unroll
      for (int r = 0; r < 8; ++r) {
        const int gm0 = row0 + r + 8 * hi;   // global row, first A tile
        const int gm1 = gm0 + 16;            // global row, second A tile
        if (col == (gm0 ^ 1)) sPos[r + 8 * hi]       = c0[r] * INV_TAU;
        if (col == (gm1 ^ 1)) sPos[16 + r + 8 * hi]  = c1[r] * INV_TAU;
        if (col == gm0)       sDiag[r + 8 * hi]      = exp2f(c0[r] * LOG2E_X2);
        if (col == gm1)       sDiag[16 + r + 8 * hi] = exp2f(c1[r] * LOG2E_X2);
      }
    }
  }

  // Sum across the 16 lanes of each half-wave (same rows, disjoint cols).
#pragma unroll
  for (int msk = 1; msk < 16; msk <<= 1) {
#pragma unroll
    for (int r = 0; r < 8; ++r) {
      s0[r] += __shfl_xor(s0[r], msk, 32);
      s1[r] += __shfl_xor(s1[r], msk, 32);
    }
  }
  if (l16 == 0) {
#pragma unroll
    for (int r = 0; r < 8; ++r) {
      sS[w][r + 8 * hi]      = s0[r];
      sS[w][16 + r + 8 * hi] = s1[r];
    }
  }
  __syncthreads();

  // Wave 0: merge the 8 waves' sums, mask self, finish lse - pos, reduce rows.
  if (tid < 32) {
    const int row = tid;
    float S = 0.0f;
#pragma unroll
    for (int ww = 0; ww < 8; ++ww) S += sS[ww][row];
    float li = logf(S - sDiag[row]) - sPos[row];
#pragma unroll
    for (int msk = 1; msk < 32; msk <<= 1) li += __shfl_xor(li, msk, 32);
    if (tid == 0) partial[blockIdx.x] = li;
  }
}

// ---------------------------------------------------------------------------
// Kernel 3: deterministic tree reduction of 256 per-block partials -> mean.
// ---------------------------------------------------------------------------
__global__ void __launch_bounds__(256) red_kernel(const float* __restrict__ partial,
                                                  float* __restrict__ out) {
  __shared__ float sbuf[256];
  const int tid = threadIdx.x;
  sbuf[tid] = partial[tid];
  __syncthreads();
  for (int st = 128; st >= 1; st >>= 1) {
    if (tid < st) sbuf[tid] += sbuf[tid + st];
    __syncthreads();
  }
  if (tid == 0) out[0] = sbuf[0] * (1.0f / (float)B_ROWS);
}

// ---------------------------------------------------------------------------
extern "C" void kernel_launch(void* const* d_in, const int* in_sizes, int n_in,
                              void* d_out, int out_size, void* d_ws, size_t ws_size,
                              hipStream_t stream) {
  const float* z = (const float*)d_in[0];          // [8192, 128] fp32
  float* zn      = (float*)d_ws;                   // 4 MB normalized copy
  float* partial = zn + (size_t)B_ROWS * NFEAT;    // 256 partial sums
  float* out     = (float*)d_out;                  // scalar fp32

  nrm_kernel<<<B_ROWS / 8, 256, 0, stream>>>(z, zn);
  ntx_kernel<<<B_ROWS / MROWS, 256, 0, stream>>>(zn, partial);
  red_kernel<<<1, 256, 0, stream>>>(partial, out);
}